// Reduce_88579405512820
// MI455X (gfx1250) — compile-verified
//
#include <hip/hip_runtime.h>
#include <hip/hip_bf16.h>
#include <stdint.h>

// Segment-sum scatter-add, MI455X (gfx1250, wave32).
// out[b, tgt[b,e], :] += messages[b, e, :]
// B=64, E=4096, N=512, D=128, fp32.
//
// Block = (batch b, column-quad q of 32 cols). 64 KB LDS fp32 accumulator
// [512 x 32] owned exclusively by the block -> no global atomics.
// Indices staged via global_load_async_to_lds_b128 (ASYNCcnt). Accumulate with
// ds_add_f32 (conflict-free: 32 consecutive dwords per wave). Messages read
// with non-temporal hint (streamed once). Drain LDS to HBM with
// global_store_async_from_lds_b128 th:TH_STORE_NT (write-once output).

#define B_DIM 64
#define E_DIM 4096
#define N_DIM 512
#define D_DIM 128
#define DQ 4            // column quads
#define COLS 32         // D_DIM / DQ  (== wave32 lanes)
#define THREADS 512     // 16 waves -> 4 waves/SIMD for latency hiding

__device__ __forceinline__ uint32_t lds_addr_of(const void* p) {
  // Low 32 bits of a flat pointer into the LDS aperture == LDS byte offset.
  return (uint32_t)(uintptr_t)p;
}

__device__ __forceinline__ void async_load_lds_b128(uint32_t lds_addr, const void* gaddr) {
  asm volatile("global_load_async_to_lds_b128 %0, %1, off"
               :: "v"(lds_addr), "v"(gaddr) : "memory");
}

__device__ __forceinline__ void async_store_lds_b128_nt(void* gaddr, uint32_t lds_addr) {
  asm volatile("global_store_async_from_lds_b128 %0, %1, off th:TH_STORE_NT"
               :: "v"(gaddr), "v"(lds_addr) : "memory");
}

__device__ __forceinline__ void wait_asynccnt0() {
  asm volatile("s_wait_asynccnt 0x0" ::: "memory");
}

__global__ __launch_bounds__(THREADS) void scatter_add_kernel(
    const float* __restrict__ messages,   // [B, E, D]
    const int*   __restrict__ tgt,        // [B, E]
    float*       __restrict__ out)        // [B, N, D]
{
  __shared__ float acc[N_DIM * COLS];     // 64 KB accumulator
  __shared__ int   idxs[E_DIM];           // 16 KB staged indices

  const int b = blockIdx.x >> 2;          // batch
  const int q = blockIdx.x & 3;           // column quad
  const int tid  = threadIdx.x;
  const int wave = tid >> 5;
  const int lane = tid & 31;

  // ---- Phase 0: zero the accumulator (ds_store_b128) ----
  float4* accv = (float4*)acc;
  #pragma unroll
  for (int i = tid; i < (N_DIM * COLS) / 4; i += THREADS) {
    accv[i] = make_float4(0.f, 0.f, 0.f, 0.f);
  }

  // ---- Phase 1: async-stage this batch's indices into LDS ----
  const int* tgt_b = tgt + (size_t)b * E_DIM;
  const uint32_t idx_base = lds_addr_of(&idxs[0]);
  #pragma unroll
  for (int i = tid; i < E_DIM / 4; i += THREADS) {      // 2 iterations, b128 each
    async_load_lds_b128(idx_base + (uint32_t)i * 16u, tgt_b + i * 4);
  }
  wait_asynccnt0();
  __syncthreads();

  // ---- Phase 2: scatter-accumulate into LDS ----
  // Wave handles messages; lane l owns column q*32 + l.
  // ds_add_f32 addresses: r*32 + lane -> 32 consecutive dwords -> 32 banks, no conflicts.
  const float* msg_b = messages + (size_t)b * E_DIM * D_DIM + q * COLS + lane;
  for (int e0 = wave * 4; e0 < E_DIM; e0 += (THREADS / 32) * 4) {
    float v[4];
    int   r[4];
    #pragma unroll
    for (int j = 0; j < 4; ++j) {
      const int e = e0 + j;
      // Non-temporal: 134 MB streamed once; keep it out of L2's useful set.
      v[j] = __builtin_nontemporal_load(msg_b + (size_t)e * D_DIM);
      r[j] = idxs[e];                    // wave-uniform ds_load (broadcast)
    }
    #pragma unroll
    for (int j = 0; j < 4; ++j) {
      atomicAdd(&acc[r[j] * COLS + lane], v[j]);   // ds_add_f32
    }
  }
  __syncthreads();   // compiler waits DScnt before barrier -> acc is final

  // ---- Phase 3: drain accumulator straight from LDS to HBM ----
  const uint32_t acc_base = lds_addr_of(&acc[0]);
  float* out_bq = out + (size_t)b * N_DIM * D_DIM + q * COLS;
  #pragma unroll
  for (int i = tid; i < (N_DIM * COLS) / 4; i += THREADS) {  // 8 iterations, b128 each
    const int r  = i >> 3;        // row 0..511
    const int cg = i & 7;         // 4-col group 0..7
    async_store_lds_b128_nt(out_bq + (size_t)r * D_DIM + cg * 4,
                            acc_base + (uint32_t)(r * COLS + cg * 4) * 4u);
  }
  wait_asynccnt0();
}

extern "C" void kernel_launch(void* const* d_in, const int* in_sizes, int n_in,
                              void* d_out, int out_size, void* d_ws, size_t ws_size,
                              hipStream_t stream) {
  (void)in_sizes; (void)n_in; (void)d_ws; (void)ws_size; (void)out_size;
  const float* messages = (const float*)d_in[0];   // [64, 4096, 128] fp32
  const int*   tgt      = (const int*)d_in[1];     // [64, 4096] int32
  // d_in[2] (atom_features_ref) only defines N; unused.
  float* out = (float*)d_out;                      // [64, 512, 128] fp32

  dim3 grid(B_DIM * DQ);     // 256 blocks: (b, q)
  dim3 block(THREADS);       // 16 waves (wave32)
  scatter_add_kernel<<<grid, block, 0, stream>>>(messages, tgt, out);
}